// MinibatchDisc_22703197126870
// MI455X (gfx1250) — compile-verified
//
#include <hip/hip_runtime.h>
#include <hip/hip_bf16.h>

// Problem constants from the reference: N=256, IN=1024, B=128, C=16.
#define NN   256
#define INF_ 1024
#define BB   128
#define CC   16
#define KDIM 1024
#define NOUT 2048          // B*C, columns of the GEMM
#define OUTW (INF_ + BB)   // 1152

typedef __attribute__((ext_vector_type(2))) float v2f;
typedef __attribute__((ext_vector_type(8))) float v8f;

// ---------------------------------------------------------------------------
// Kernel 1: mat[n, b*16+c] = sum_i x[n,i] * T[i, b*16+c]
// f32 WMMA (V_WMMA_F32_16X16X4_F32), one wave per 16x16 output tile.
// Grid: 512 blocks x 128 threads (4 waves/block). 2048 tiles total.
// Waves in a block share the same A rows (same tm, consecutive tn).
// ---------------------------------------------------------------------------
__global__ __launch_bounds__(128) void gemm_wmma_f32(
    const float* __restrict__ x,   // [256, 1024]
    const float* __restrict__ T,   // [1024, 2048]
    float* __restrict__ mat)       // [256, 2048]
{
    const int lane = threadIdx.x & 31;
    const int wave = threadIdx.x >> 5;
    const int tile = blockIdx.x * 4 + wave;   // 0..2047
    const int tn = tile & 127;                // n-tile (fastest)
    const int tm = tile >> 7;                 // m-tile 0..15

    const int row0 = tm << 4;
    const int col0 = tn << 4;
    const int half = lane >> 4;               // selects K-pair (0: k0,k1  1: k2,k3)
    const int lrow = lane & 15;               // A row / B+D column within tile

    // A: lane reads x[row0+lrow, k + 2*half + {0,1}]  (8B contiguous)
    const float* __restrict__ Aptr = x + (size_t)(row0 + lrow) * KDIM + (half << 1);
    // B: lane reads T[k + 2*half + {0,1}, col0+lrow]  (two strided b32)
    const float* __restrict__ Bptr = T + (size_t)(half << 1) * NOUT + (col0 + lrow);

    v8f acc = {};
#pragma unroll 4
    for (int k = 0; k < KDIM; k += 4) {
        v2f a = *(const v2f*)(Aptr + k);
        v2f b;
        b.x = Bptr[(size_t)k * NOUT];
        b.y = Bptr[(size_t)k * NOUT + NOUT];
        // D = A(16x4) * B(4x16) + C, full f32 — matches reference dtype.
        acc = __builtin_amdgcn_wmma_f32_16x16x4_f32(
            /*neg_a=*/false, a, /*neg_b=*/false, b,
            /*c_mod=*/(short)0, acc, /*reuse_a=*/false, /*reuse_b=*/false);
    }

    // D layout: VGPR r, lane l -> D[r + 8*(l>=16), l&15]
    float* __restrict__ D = mat + (size_t)(row0 + (half << 3)) * NOUT + col0 + lrow;
#pragma unroll
    for (int r = 0; r < 8; ++r)
        D[(size_t)r * NOUT] = acc[r];
}

// ---------------------------------------------------------------------------
// Kernel 2: per (n,b): o = sum_j exp(-sum_c |mat[n,b,c]-mat[j,b,c]|)
// Also copies x row n into out[n, 0:1024]. One block per n, thread = b.
// All mat[j] reads are float4, coalesced across b (consecutive 64B chunks).
// ---------------------------------------------------------------------------
__global__ __launch_bounds__(128) void l1exp_kernel(
    const float* __restrict__ x,    // [256, 1024]
    const float* __restrict__ mat,  // [256, 2048]
    float* __restrict__ out)        // [256, 1152]
{
    const int n = blockIdx.x;       // 0..255
    const int b = threadIdx.x;      // 0..127

    // Copy flat part: x[n,:] -> out[n, 0:1024] (two float4 per thread).
    const float4* __restrict__ x4 = (const float4*)(x + (size_t)n * INF_);
    float4* __restrict__ o4 = (float4*)(out + (size_t)n * OUTW);
    o4[b]       = x4[b];
    o4[b + 128] = x4[b + 128];

    // Own row mat[n, b, :] in registers.
    float m[CC];
    const float4* __restrict__ mr = (const float4*)(mat + (size_t)n * NOUT + b * CC);
    {
        float4 r0 = mr[0], r1 = mr[1], r2 = mr[2], r3 = mr[3];
        m[0]=r0.x; m[1]=r0.y; m[2]=r0.z; m[3]=r0.w;
        m[4]=r1.x; m[5]=r1.y; m[6]=r1.z; m[7]=r1.w;
        m[8]=r2.x; m[9]=r2.y; m[10]=r2.z; m[11]=r2.w;
        m[12]=r3.x; m[13]=r3.y; m[14]=r3.z; m[15]=r3.w;
    }

    float acc = 0.0f;
#pragma unroll 2
    for (int j = 0; j < NN; ++j) {
        const float4* __restrict__ q4 = (const float4*)(mat + (size_t)j * NOUT + b * CC);
        float4 q0 = q4[0], q1 = q4[1], q2 = q4[2], q3 = q4[3];
        float d =
            fabsf(m[0]-q0.x) + fabsf(m[1]-q0.y) + fabsf(m[2]-q0.z) + fabsf(m[3]-q0.w) +
            fabsf(m[4]-q1.x) + fabsf(m[5]-q1.y) + fabsf(m[6]-q1.z) + fabsf(m[7]-q1.w) +
            fabsf(m[8]-q2.x) + fabsf(m[9]-q2.y) + fabsf(m[10]-q2.z) + fabsf(m[11]-q2.w) +
            fabsf(m[12]-q3.x) + fabsf(m[13]-q3.y) + fabsf(m[14]-q3.z) + fabsf(m[15]-q3.w);
        acc += __expf(-d);   // native v_exp_f32; includes self term exp(0)=1
    }

    out[(size_t)n * OUTW + INF_ + b] = acc;
}

extern "C" void kernel_launch(void* const* d_in, const int* in_sizes, int n_in,
                              void* d_out, int out_size, void* d_ws, size_t ws_size,
                              hipStream_t stream) {
    (void)in_sizes; (void)n_in; (void)out_size; (void)ws_size;
    const float* x = (const float*)d_in[0];   // [256, 1024] f32
    const float* T = (const float*)d_in[1];   // [1024, 128, 16] f32 == [1024, 2048]
    float* out = (float*)d_out;               // [256, 1152] f32
    float* mat = (float*)d_ws;                // [256, 2048] f32 = 2 MB scratch

    // GEMM: 2048 tiles of 16x16, 4 waves/block -> 512 blocks of 128 threads.
    gemm_wmma_f32<<<512, 128, 0, stream>>>(x, T, mat);
    // Pairwise L1 + exp reduction, fused with the flat-copy.
    l1exp_kernel<<<NN, BB, 0, stream>>>(x, mat, out);
}